// SEGPostProcessor_34935263986230
// MI455X (gfx1250) — compile-verified
//
#include <hip/hip_runtime.h>

// SEG binarize with multiple thresholds:
//   in : pred [32, 1, 1024, 1024] fp32
//   out: bitmap [32, 4, 1024, 1024] fp32,  out[n,t,p] = pred[n,p] > thr[t]
//
// Pure streaming kernel: 4B read + 16B written per element (~0.0125 FLOP/B).
// Bandwidth-bound on MI455X (23.3 TB/s -> ~27us floor). Strategy:
//  - one global_load_b128 (TH=NT) per thread: 4 pred elements
//  - four global_store_b128 (TH=NT) per thread, one per threshold plane.
//    Output (512 MiB) is write-once and > 192 MB L2 -> NT avoids thrashing.
//  - store base biased to the midpoint of the 4 planes (+0x600000 B) so all
//    four stores share ONE VGPR address pair with signed-24-bit immediate
//    offsets {-0x600000, -0x200000, +0x200000, +0x600000}: no per-plane
//    64-bit VALU address chains.

typedef float v4f __attribute__((ext_vector_type(4)));

#define HW_BITS   20u                 // H*W = 1024*1024 = 2^20 elements/plane
#define HW_MASK   ((1u << HW_BITS) - 1u)
#define NTHRESH   4
// midpoint bias in FLOATS: 1.5 planes = 3 * 2^19 elements (= 0x600000 bytes)
#define MID_BIAS  (3u << 19)

__global__ __launch_bounds__(256) void seg_binarize_mt_kernel(
    const float* __restrict__ pred,
    float* __restrict__ out,
    unsigned int n_elems)   // total pred elements (N*H*W)
{
    unsigned int i = (blockIdx.x * blockDim.x + threadIdx.x) * 4u;
    if (i >= n_elems) return;

    // 128-bit NT load of 4 consecutive pred values (16B aligned: i % 4 == 0)
    v4f v = __builtin_nontemporal_load((const v4f*)(pred + i));

    unsigned int n = i >> HW_BITS;          // batch index
    unsigned int p = i & HW_MASK;           // pixel index within plane
    // mid = &out[n][1.5][p] : midpoint of this batch's 4 threshold planes
    float* mid = out + (((size_t)n * NTHRESH) << HW_BITS) + p + MID_BIAS;

    const float thr[NTHRESH] = {0.2f, 0.3f, 0.4f, 0.5f};

#pragma unroll
    for (int t = 0; t < NTHRESH; ++t) {
        v4f b;
        b.x = (v.x > thr[t]) ? 1.0f : 0.0f;
        b.y = (v.y > thr[t]) ? 1.0f : 0.0f;
        b.z = (v.z > thr[t]) ? 1.0f : 0.0f;
        b.w = (v.w > thr[t]) ? 1.0f : 0.0f;
        // compile-time float offset in [-0x180000, +0x180000] elements
        // -> byte IOFFSET in {-0x600000,-0x200000,+0x200000,+0x600000},
        //    all within the signed 24-bit global_store immediate.
        ptrdiff_t off = ((ptrdiff_t)t << HW_BITS) - (ptrdiff_t)MID_BIAS;
        __builtin_nontemporal_store(b, (v4f*)(mid + off));
    }
}

extern "C" void kernel_launch(void* const* d_in, const int* in_sizes, int n_in,
                              void* d_out, int out_size, void* d_ws, size_t ws_size,
                              hipStream_t stream) {
    (void)n_in; (void)out_size; (void)d_ws; (void)ws_size;

    const float* pred = (const float*)d_in[0];
    float* out = (float*)d_out;
    unsigned int n_elems = (unsigned int)in_sizes[0];   // 32 * 1024 * 1024

    const unsigned int threads_needed = n_elems / 4u;   // one float4 per thread
    const unsigned int block = 256u;
    const unsigned int grid = (threads_needed + block - 1u) / block;

    seg_binarize_mt_kernel<<<grid, block, 0, stream>>>(pred, out, n_elems);
}